// TopKActivation_26551487824726
// MI455X (gfx1250) — compile-verified
//
#include <hip/hip_runtime.h>

#define COLS 24576   // row length (elements)
#define KSEL 64      // top-k
#define TPB  1024    // 32 waves (wave32) per workgroup

typedef __attribute__((ext_vector_type(4))) unsigned int u32x4;
typedef __attribute__((ext_vector_type(8))) int          i32x8;
typedef __attribute__((ext_vector_type(4))) int          i32x4;

// Order-preserving float32 -> uint32 key: larger float => larger key.
__device__ __forceinline__ unsigned int f2key(float f) {
  unsigned int u = __float_as_uint(f);
  return (u & 0x80000000u) ? ~u : (u | 0x80000000u);
}

__global__ void __launch_bounds__(TPB)
topk_scatter_kernel(const float* __restrict__ z, float* __restrict__ out) {
  extern __shared__ float srow[];          // COLS floats = 96 KB staged row
  __shared__ unsigned int hist[256];       // radix histogram
  __shared__ unsigned int s_prefix;        // resolved high bits of k-th key
  __shared__ unsigned int s_krem;          // remaining rank within prefix

  const unsigned tid = threadIdx.x;
  const unsigned row = blockIdx.x;
  const float* zrow = z + (size_t)row * COLS;
  float*       orow = out + (size_t)row * COLS;

  // ---- Stage the entire row into LDS with ONE TDM descriptor (wave 0) ----
  if (tid < 32u) {
    unsigned long long ga = (unsigned long long)(uintptr_t)zrow;
    // Low 32 bits of the flat LDS-aperture address == LDS byte offset.
    unsigned int lds_off = (unsigned int)(uintptr_t)(void*)srow;

    u32x4 g0;
    g0[0] = 1u;                                    // count=1 (valid), user mode
    g0[1] = lds_off;                               // lds_addr (bytes)
    g0[2] = (unsigned int)ga;                      // global_addr[31:0]
    g0[3] = (unsigned int)((ga >> 32) & 0x01FFFFFFull) | 0x80000000u; // addr[56:32] | type=2

    i32x8 g1;
    g1[0] = (int)(2u << 16);                       // wg_mask=0, data_size=2 (4 bytes)
    g1[1] = (int)((COLS & 0xFFFFu) << 16);         // tensor_dim0[15:0] (atomic_barrier_addr=0)
    g1[2] = (int)(((COLS >> 16) & 0xFFFFu) | (1u << 16)); // tensor_dim0[31:16]; tensor_dim1=1
    g1[3] = (int)((COLS & 0xFFFFu) << 16);         // tensor_dim1 hi=0; tile_dim0=COLS
    g1[4] = 1;                                     // tile_dim1=1; tile_dim2=0
    g1[5] = COLS;                                  // tensor_dim0_stride
    g1[6] = 0;
    g1[7] = 0;

    i32x4 zero4 = {0, 0, 0, 0};
    i32x8 zero8 = {0, 0, 0, 0, 0, 0, 0, 0};
    __builtin_amdgcn_tensor_load_to_lds(g0, g1, zero4, zero4, zero8, 0);
    __builtin_amdgcn_s_wait_tensorcnt(0);          // wave0's TENSORcnt -> 0: row is in LDS
  }
  if (tid == 0u) { s_prefix = 0u; s_krem = KSEL; }
  __syncthreads();                                 // row visible to all 32 waves

  // ---- Radix-select the K-th largest key: 4 passes of 8 bits, MSB first ----
  const float4* srow4 = (const float4*)srow;
#pragma unroll 1
  for (int pass = 0; pass < 4; ++pass) {
    const int shift = 24 - 8 * pass;
    if (tid < 256u) hist[tid] = 0u;
    __syncthreads();

    const unsigned prefix = s_prefix;
    const unsigned pmask  = (pass == 0) ? 0u : (0xFFFFFFFFu << (shift + 8));

    for (unsigned i = tid; i < COLS / 4; i += TPB) {
      float4 v = srow4[i];
      unsigned k0 = f2key(v.x), k1 = f2key(v.y), k2 = f2key(v.z), k3 = f2key(v.w);
      if ((k0 & pmask) == prefix) atomicAdd(&hist[(k0 >> shift) & 0xFFu], 1u);
      if ((k1 & pmask) == prefix) atomicAdd(&hist[(k1 >> shift) & 0xFFu], 1u);
      if ((k2 & pmask) == prefix) atomicAdd(&hist[(k2 >> shift) & 0xFFu], 1u);
      if ((k3 & pmask) == prefix) atomicAdd(&hist[(k3 >> shift) & 0xFFu], 1u);
    }
    __syncthreads();

    if (tid == 0u) {
      unsigned krem = s_krem, acc = 0u;
      int b = 255;
      for (; b > 0; --b) {
        unsigned c = hist[b];
        if (acc + c >= krem) break;
        acc += c;
      }
      s_krem   = krem - acc;
      s_prefix = prefix | ((unsigned)b << shift);
    }
    __syncthreads();
  }

  // ---- Scatter: keep values whose key >= k-th largest key, else zero ----
  const unsigned thr = s_prefix;
  float4* orow4 = (float4*)orow;
  for (unsigned i = tid; i < COLS / 4; i += TPB) {
    float4 v = srow4[i];
    v.x = (f2key(v.x) >= thr) ? v.x : 0.0f;
    v.y = (f2key(v.y) >= thr) ? v.y : 0.0f;
    v.z = (f2key(v.z) >= thr) ? v.z : 0.0f;
    v.w = (f2key(v.w) >= thr) ? v.w : 0.0f;
    orow4[i] = v;                                  // global_store_b128
  }
}

extern "C" void kernel_launch(void* const* d_in, const int* in_sizes, int n_in,
                              void* d_out, int out_size, void* d_ws, size_t ws_size,
                              hipStream_t stream) {
  const float* z  = (const float*)d_in[0];
  float* out      = (float*)d_out;
  const int rows  = in_sizes[0] / COLS;            // 8192
  const size_t shmem = (size_t)COLS * sizeof(float); // 96 KB dynamic LDS

  // Allow >64KB dynamic LDS (no-op if already permitted); not a stream op,
  // safe under graph capture and deterministic.
  (void)hipFuncSetAttribute((const void*)topk_scatter_kernel,
                            hipFuncAttributeMaxDynamicSharedMemorySize,
                            (int)shmem);

  topk_scatter_kernel<<<rows, TPB, shmem, stream>>>(z, out);
}